// TGNN_34883724378623
// MI455X (gfx1250) — compile-verified
//
#include <hip/hip_runtime.h>
#include <hip/hip_bf16.h>

typedef __attribute__((ext_vector_type(2))) float v2f;
typedef __attribute__((ext_vector_type(8))) float v8f;

#define N_NODES   50000
#define N_EDGES   800000
#define EMB       128
#define TIME_DIM  64
#define NEG_SLOPE 0.2f

// ---------------------------------------------------------------------------
// 1) Embedding gather (also the residual kept to the end): res[n,:] = emb[x[n],:]
// ---------------------------------------------------------------------------
__global__ void k_gather_emb(const int* __restrict__ x, const float* __restrict__ emb,
                             float* __restrict__ res) {
    int i = blockIdx.x * blockDim.x + threadIdx.x;      // over N * (EMB/4)
    int node = i >> 5;                                  // EMB/4 == 32
    int c4   = i & 31;
    if (node >= N_NODES) return;
    const float4* s = (const float4*)(emb + (size_t)x[node] * EMB);
    ((float4*)(res + (size_t)node * EMB))[c4] = s[c4];
}

// ---------------------------------------------------------------------------
// 2) Dense GEMM C[N_NODES, NCOL] = A[N_NODES, K] * W[NCOL, K]^T using
//    V_WMMA_F32_16X16X4_F32.  One 16x16 output tile per wave, 8 waves/block.
//    A-frag (16x4): lane m = lane&15, kb = 2*(lane>>4) -> {A[m][kb], A[m][kb+1]}
//    B-frag (4x16): lane n = lane&15, same kb          -> {W[n][kb], W[n][kb+1]}
//    C/D: lanes 0-15 hold M=v, lanes 16-31 hold M=v+8, N = lane&15.
// ---------------------------------------------------------------------------
template <int K, int NCOL>
__global__ void k_gemm_wmma(const float* __restrict__ A, const float* __restrict__ W,
                            float* __restrict__ C) {
    constexpr int NT = NCOL / 16;
    const int wave = threadIdx.x >> 5;
    const int lane = threadIdx.x & 31;
    const int tile = blockIdx.x * (blockDim.x >> 5) + wave;
    if (tile >= (N_NODES / 16) * NT) return;            // whole-wave uniform
    const int tm = tile / NT;
    const int tn = tile % NT;

    const int r  = lane & 15;
    const int kb = (lane >> 4) << 1;                    // 0 or 2
    const v2f* a2 = (const v2f*)(A + (size_t)(tm * 16 + r) * K + kb);
    const v2f* b2 = (const v2f*)(W + (size_t)(tn * 16 + r) * K + kb);

    v8f acc = {0.f, 0.f, 0.f, 0.f, 0.f, 0.f, 0.f, 0.f};
    #pragma unroll 8
    for (int k2 = 0; k2 < K / 2; k2 += 2) {             // 4 K-values per step
        v2f va = a2[k2];
        v2f vb = b2[k2];
        acc = __builtin_amdgcn_wmma_f32_16x16x4_f32(
                  false, va, false, vb, (short)0, acc, false, false);
    }

    const int mbase = (lane >> 4) * 8;
    float* crow = C + (size_t)(tm * 16 + mbase) * NCOL + tn * 16 + (lane & 15);
    #pragma unroll
    for (int v = 0; v < 8; ++v)
        crow[(size_t)v * NCOL] = acc[v];
}

// ---------------------------------------------------------------------------
// 3) Per-node attention scalars: sA[n] = G[n,:]·a_s, sD[n] = G[n,:]·a_d
// ---------------------------------------------------------------------------
template <int C>
__global__ void k_rowdot2(const float* __restrict__ G, const float* __restrict__ a_s,
                          const float* __restrict__ a_d,
                          float* __restrict__ sA, float* __restrict__ sD) {
    const int node = blockIdx.x * (blockDim.x >> 5) + (threadIdx.x >> 5);
    const int lane = threadIdx.x & 31;
    if (node >= N_NODES) return;
    const float* row = G + (size_t)node * C;
    float s0 = 0.f, s1 = 0.f;
    #pragma unroll
    for (int c = lane; c < C; c += 32) {
        float v = row[c];
        s0 += v * a_s[c];
        s1 += v * a_d[c];
    }
    #pragma unroll
    for (int off = 16; off; off >>= 1) {
        s0 += __shfl_xor(s0, off, 32);
        s1 += __shfl_xor(s1, off, 32);
    }
    if (lane == 0) { sA[node] = s0; sD[node] = s1; }
}

// ---------------------------------------------------------------------------
// 4) Time-feature collapse: a_time(e) = c1*edge_time[e] + c0 with
//    c1 = te_w·a_t, c0 = te_b·a_t.  Also zeroes the alpha-sum accumulator.
// ---------------------------------------------------------------------------
__global__ void k_time_consts(const float* __restrict__ te_w, const float* __restrict__ te_b,
                              const float* __restrict__ a_t, float* __restrict__ cbuf) {
    if (threadIdx.x == 0 && blockIdx.x == 0) {
        float c1 = 0.f, c0 = 0.f;
        for (int j = 0; j < TIME_DIM; ++j) {
            c1 += te_w[j] * a_t[j];
            c0 += te_b[j] * a_t[j];
        }
        cbuf[0] = c0;
        cbuf[1] = c1;
        cbuf[2] = 0.f;                                  // esum accumulator
    }
}

// ---------------------------------------------------------------------------
// 5) Per-edge alpha = exp(-leaky((sA[src]+sD[dst]+c1*t+c0)*inv_sqrt)) and
//    global sum of alpha (block reduce + one atomic per block).
// ---------------------------------------------------------------------------
__global__ void k_edge_alpha(const int* __restrict__ src, const int* __restrict__ dst,
                             const float* __restrict__ et,
                             const float* __restrict__ sA, const float* __restrict__ sD,
                             const float* __restrict__ cbuf, float inv_sqrt,
                             float* __restrict__ alpha, float* __restrict__ esum) {
    const int e = blockIdx.x * blockDim.x + threadIdx.x;
    float v = 0.f;
    if (e < N_EDGES) {
        float a = (sA[src[e]] + sD[dst[e]] + cbuf[1] * et[e] + cbuf[0]) * inv_sqrt;
        a = (a >= 0.f) ? a : NEG_SLOPE * a;
        v = __expf(-a);
        alpha[e] = v;
    }
    __shared__ float red[8];
    #pragma unroll
    for (int off = 16; off; off >>= 1) v += __shfl_xor(v, off, 32);
    if ((threadIdx.x & 31) == 0) red[threadIdx.x >> 5] = v;
    __syncthreads();
    if (threadIdx.x < 8) {
        float t = red[threadIdx.x];
        #pragma unroll
        for (int off = 4; off; off >>= 1) t += __shfl_xor(t, off, 32);
        if (threadIdx.x == 0) atomicAdd(esum, t);
    }
}

// ---------------------------------------------------------------------------
// 6) Zero-fill
// ---------------------------------------------------------------------------
__global__ void k_zero(float* __restrict__ p, size_t n) {
    size_t i = (size_t)blockIdx.x * blockDim.x + threadIdx.x;
    if (i < n) p[i] = 0.f;
}

// ---------------------------------------------------------------------------
// 7) Edge scatter: out[dst[e],c] += G[src[e],c] * alpha[e]/esum.
//    Channel-major indexing: a full wave shares one edge -> scalar src/dst/alpha.
// ---------------------------------------------------------------------------
template <int C>
__global__ void k_scatter(const int* __restrict__ src, const int* __restrict__ dst,
                          const float* __restrict__ alpha, const float* __restrict__ esum,
                          const float* __restrict__ G, float* __restrict__ out) {
    constexpr int SHIFT = (C == 256) ? 8 : 7;
    const size_t idx = (size_t)blockIdx.x * blockDim.x + threadIdx.x;
    if (idx >= (size_t)N_EDGES * C) return;
    const int e = (int)(idx >> SHIFT);
    const int c = (int)(idx & (C - 1));
    const float w = alpha[e] * (1.0f / esum[0]);
    const float v = G[(size_t)src[e] * C + c] * w;
    atomicAdd(out + (size_t)dst[e] * C + c, v);
}

// ---------------------------------------------------------------------------
// 8) Elementwise: in-place leaky ReLU; final leaky + residual.
// ---------------------------------------------------------------------------
__global__ void k_leaky(float* __restrict__ b, size_t n) {
    size_t i = (size_t)blockIdx.x * blockDim.x + threadIdx.x;
    if (i < n) {
        float v = b[i];
        b[i] = (v >= 0.f) ? v : NEG_SLOPE * v;
    }
}

__global__ void k_final(float* __restrict__ out, const float* __restrict__ res) {
    size_t i = (size_t)blockIdx.x * blockDim.x + threadIdx.x;
    if (i < (size_t)N_NODES * EMB) {
        float v = out[i];
        v = (v >= 0.f) ? v : NEG_SLOPE * v;
        out[i] = v + res[i];
    }
}

// ---------------------------------------------------------------------------
extern "C" void kernel_launch(void* const* d_in, const int* in_sizes, int n_in,
                              void* d_out, int out_size, void* d_ws, size_t ws_size,
                              hipStream_t stream) {
    const int*   x    = (const int*)d_in[0];
    const int*   src  = (const int*)d_in[1];
    const int*   dst  = src + N_EDGES;
    const float* et   = (const float*)d_in[2];
    const float* emb  = (const float*)d_in[3];
    const float* lin0 = (const float*)d_in[4];
    const float* te0w = (const float*)d_in[5];
    const float* te0b = (const float*)d_in[6];
    const float* as0  = (const float*)d_in[7];
    const float* ad0  = (const float*)d_in[8];
    const float* at0  = (const float*)d_in[9];
    const float* lin1 = (const float*)d_in[10];
    const float* te1w = (const float*)d_in[11];
    const float* te1b = (const float*)d_in[12];
    const float* as1  = (const float*)d_in[13];
    const float* ad1  = (const float*)d_in[14];
    const float* at1  = (const float*)d_in[15];
    float* out = (float*)d_out;

    // Workspace layout (floats)
    float* ws    = (float*)d_ws;
    float* res   = ws;                                   // N*128
    float* gbuf  = res  + (size_t)N_NODES * 128;         // N*256 (GEMM out, both layers)
    float* obuf  = gbuf + (size_t)N_NODES * 256;         // N*256 (aggregation)
    float* sA    = obuf + (size_t)N_NODES * 256;         // N
    float* sD    = sA   + N_NODES;                       // N
    float* alpha = sD   + N_NODES;                       // E
    float* cbuf  = alpha + N_EDGES;                      // [c0, c1, esum]

    const int T = 256;
    const float inv16  = 1.0f / 16.0f;                   // 1/sqrt(256)
    const float inv128 = 0.08838834764831845f;           // 1/sqrt(128)

    // h = emb[x]
    k_gather_emb<<<(N_NODES * 32 + T - 1) / T, T, 0, stream>>>(x, emb, res);

    // ---- layer 0 ----
    k_gemm_wmma<128, 256><<<(N_NODES / 16) * 16 / 8, T, 0, stream>>>(res, lin0, gbuf);
    k_rowdot2<256><<<(N_NODES + 7) / 8, T, 0, stream>>>(gbuf, as0, ad0, sA, sD);
    k_time_consts<<<1, 64, 0, stream>>>(te0w, te0b, at0, cbuf);
    k_edge_alpha<<<(N_EDGES + T - 1) / T, T, 0, stream>>>(src, dst, et, sA, sD, cbuf,
                                                          inv16, alpha, cbuf + 2);
    k_zero<<<((size_t)N_NODES * 256 + T - 1) / T, T, 0, stream>>>(obuf, (size_t)N_NODES * 256);
    k_scatter<256><<<((size_t)N_EDGES * 256 + T - 1) / T, T, 0, stream>>>(src, dst, alpha,
                                                                          cbuf + 2, gbuf, obuf);
    k_leaky<<<((size_t)N_NODES * 256 + T - 1) / T, T, 0, stream>>>(obuf, (size_t)N_NODES * 256);

    // ---- layer 1 ----
    k_gemm_wmma<256, 128><<<(N_NODES / 16) * 8 / 8, T, 0, stream>>>(obuf, lin1, gbuf);
    k_rowdot2<128><<<(N_NODES + 7) / 8, T, 0, stream>>>(gbuf, as1, ad1, sA, sD);
    k_time_consts<<<1, 64, 0, stream>>>(te1w, te1b, at1, cbuf);
    k_edge_alpha<<<(N_EDGES + T - 1) / T, T, 0, stream>>>(src, dst, et, sA, sD, cbuf,
                                                          inv128, alpha, cbuf + 2);
    k_zero<<<((size_t)N_NODES * 128 + T - 1) / T, T, 0, stream>>>(out, (size_t)N_NODES * 128);
    k_scatter<128><<<((size_t)N_EDGES * 128 + T - 1) / T, T, 0, stream>>>(src, dst, alpha,
                                                                          cbuf + 2, gbuf, out);
    k_final<<<((size_t)N_NODES * 128 + T - 1) / T, T, 0, stream>>>(out, res);
}